// RegressionIOULoss_16312285790561
// MI455X (gfx1250) — compile-verified
//
#include <hip/hip_runtime.h>
#include <hip/hip_bf16.h>
#include <math.h>

#define TILE 256
#define POS_LB 0.55f

// ---- LDS byte-offset of a shared-memory pointer (addrspace(3) ptrtoint) ----
__device__ __forceinline__ unsigned lds_off(const void* p) {
  return (unsigned)(unsigned long long)(const __attribute__((address_space(3))) char*)p;
}

// ---- async global -> LDS, one dword per active lane (GVS mode, CDNA5) ----
__device__ __forceinline__ void async_load_dw(unsigned lds_byte, unsigned gbl_byte_off,
                                              const float* base) {
  asm volatile("global_load_async_to_lds_b32 %0, %1, %2"
               :: "v"(lds_byte), "v"(gbl_byte_off), "s"(base) : "memory");
}
__device__ __forceinline__ void wait_async0() {
  asm volatile("s_wait_asynccnt 0" ::: "memory");
}

// ---- rotated rectangle corners (matches reference box2corners ordering, CCW) ----
__device__ __forceinline__ void box_corners(float x, float y, float w, float l, float a,
                                            float* cx, float* cy) {
  float s, c;
  __sincosf(a, &s, &c);
  const float tx[4] = {0.5f, -0.5f, -0.5f, 0.5f};
  const float ty[4] = {0.5f, 0.5f, -0.5f, -0.5f};
#pragma unroll
  for (int i = 0; i < 4; ++i) {
    float px = w * tx[i], py = l * ty[i];
    cx[i] = px * c - py * s + x;
    cy[i] = px * s + py * c + y;
  }
}

// ---- convex intersection area via Sutherland-Hodgman clip of c1 by c2 ----
__device__ float rect_intersection_area(const float* c1x, const float* c1y,
                                        const float* c2x, const float* c2y) {
  float px[8], py[8];
#pragma unroll
  for (int i = 0; i < 4; ++i) { px[i] = c1x[i]; py[i] = c1y[i]; }
  int n = 4;
#pragma unroll
  for (int e = 0; e < 4; ++e) {
    float ax = c2x[e], ay = c2y[e];
    float ex = c2x[(e + 1) & 3] - ax;
    float ey = c2y[(e + 1) & 3] - ay;
    float qx[8] = {0}, qy[8] = {0};
    int m = 0;
    for (int i = 0; i < n; ++i) {
      int ip = (i + 1 == n) ? 0 : i + 1;
      float dc = ex * (py[i] - ay) - ey * (px[i] - ax);
      float dn = ex * (py[ip] - ay) - ey * (px[ip] - ax);
      if (dc >= 0.f) { qx[m] = px[i]; qy[m] = py[i]; ++m; }
      if (dc * dn < 0.f) {
        float t = dc / (dc - dn);
        qx[m] = px[i] + t * (px[ip] - px[i]);
        qy[m] = py[i] + t * (py[ip] - py[i]);
        ++m;
      }
    }
    n = m;
#pragma unroll
    for (int i = 0; i < 8; ++i) { px[i] = qx[i]; py[i] = qy[i]; }
    if (n == 0) break;
  }
  float area = 0.f;
  for (int i = 0; i < n; ++i) {
    int ip = (i + 1 == n) ? 0 : i + 1;
    area += px[i] * py[ip] - px[ip] * py[i];
  }
  return 0.5f * fabsf(area);
}

// ---- smallest enclosing rectangle area over 28 candidate directions ----
__device__ float enclosing_area(const float* c1x, const float* c1y,
                                const float* c2x, const float* c2y) {
  float ptx[8], pty[8];
#pragma unroll
  for (int i = 0; i < 4; ++i) {
    ptx[i] = c1x[i];     pty[i] = c1y[i];
    ptx[i + 4] = c2x[i]; pty[i + 4] = c2y[i];
  }
  float best = 1e18f;
#pragma unroll
  for (int i = 0; i < 8; ++i) {
#pragma unroll
    for (int j = i + 1; j < 8; ++j) {
      float dx = ptx[j] - ptx[i], dy = pty[j] - pty[i];
      float nrm = __builtin_sqrtf(dx * dx + dy * dy);
      float inv = 1.0f / fmaxf(nrm, 1e-6f);
      float ux = dx * inv, uy = dy * inv;
      float pumin = 1e30f, pumax = -1e30f, pnmin = 1e30f, pnmax = -1e30f;
#pragma unroll
      for (int p = 0; p < 8; ++p) {
        float pu =  ux * ptx[p] + uy * pty[p];
        float pn = -uy * ptx[p] + ux * pty[p];
        pumin = fminf(pumin, pu); pumax = fmaxf(pumax, pu);
        pnmin = fminf(pnmin, pn); pnmax = fmaxf(pnmax, pn);
      }
      float area = (pumax - pumin) * (pnmax - pnmin);
      best = fminf(best, (nrm > 1e-6f) ? area : 1e18f);
    }
  }
  return best;
}

__global__ void __launch_bounds__(TILE)
giou3d_kernel(const float* __restrict__ pred, const float* __restrict__ targ,
              const float* __restrict__ iou, float* __restrict__ acc, int N) {
  __shared__ float sp[TILE * 7];
  __shared__ float st[TILE * 7];
  __shared__ float si[TILE];

  const int tid  = threadIdx.x;
  const int base = blockIdx.x * TILE;

  // ---- stage AoS tile into LDS with coalesced async B32 loads ----
  if (base + TILE <= N) {
    // Full tile (the common case): uniform, no per-element guards, 32-bit math.
    const unsigned eb4 = (unsigned)base * 28u;  // base*7*4 bytes
#pragma unroll
    for (int k = 0; k < 7; ++k) {
      const unsigned fi = (unsigned)(k * TILE + tid);
      const unsigned go = eb4 + fi * 4u;
      async_load_dw(lds_off(&sp[fi]), go, pred);
      async_load_dw(lds_off(&st[fi]), go, targ);
    }
    async_load_dw(lds_off(&si[tid]), (unsigned)(base + tid) * 4u, iou);
  } else {
    // Ragged tail: per-element guards.
    const unsigned rem7 = (unsigned)(N - base) * 7u;
#pragma unroll
    for (int k = 0; k < 7; ++k) {
      const unsigned fi = (unsigned)(k * TILE + tid);
      if (fi < rem7) {
        const unsigned go = (unsigned)base * 28u + fi * 4u;
        async_load_dw(lds_off(&sp[fi]), go, pred);
        async_load_dw(lds_off(&st[fi]), go, targ);
      }
    }
    if (base + tid < N) {
      async_load_dw(lds_off(&si[tid]), (unsigned)(base + tid) * 4u, iou);
    }
  }
  wait_async0();
  __syncthreads();

  float loss_masked = 0.f, mask_v = 0.f;
  if (base + tid < N) {
    const float* P = &sp[tid * 7];
    const float* T = &st[tid * 7];

    float c1x[4], c1y[4], c2x[4], c2y[4];
    box_corners(P[0], P[1], P[3], P[4], P[6], c1x, c1y);
    box_corners(T[0], T[1], T[3], T[4], T[6], c2x, c2y);

    float inter2d = rect_intersection_area(c1x, c1y, c2x, c2y);

    float zmax1 = P[2] + P[5] * 0.5f, zmin1 = P[2] - P[5] * 0.5f;
    float zmax2 = T[2] + T[5] * 0.5f, zmin2 = T[2] - T[5] * 0.5f;
    float oz = fmaxf(fminf(zmax1, zmax2) - fmaxf(zmin1, zmin2), 0.f);
    float inter3d = inter2d * oz;

    float v1 = P[3] * P[4] * P[5];
    float v2 = T[3] * T[4] * T[5];
    float uni = v1 + v2 - inter3d;
    float iou3d = inter3d / fmaxf(uni, 1e-8f);

    float zr = fmaxf(fmaxf(zmax1, zmax2) - fminf(zmin1, zmin2), 0.f);
    float vc = fmaxf(enclosing_area(c1x, c1y, c2x, c2y) * zr, 1e-8f);

    float loss = 1.f - iou3d + (vc - uni) / vc;
    float mk = (si[tid] >= POS_LB) ? 1.f : 0.f;
    loss_masked = loss * mk;
    mask_v = mk;
  }

  // ---- wave32 reduction, one fp32 atomic pair per wave ----
  float a = loss_masked, b = mask_v;
#pragma unroll
  for (int off = 16; off > 0; off >>= 1) {
    a += __shfl_down(a, off, 32);
    b += __shfl_down(b, off, 32);
  }
  if ((tid & 31) == 0) {
    __hip_atomic_fetch_add(&acc[0], a, __ATOMIC_RELAXED, __HIP_MEMORY_SCOPE_AGENT);
    __hip_atomic_fetch_add(&acc[1], b, __ATOMIC_RELAXED, __HIP_MEMORY_SCOPE_AGENT);
  }
}

__global__ void zero_acc(float* acc) { acc[0] = 0.f; acc[1] = 0.f; }

__global__ void finalize(const float* __restrict__ acc, float* __restrict__ out) {
  float cnt = acc[1];
  out[0] = (cnt > 0.f) ? (acc[0] / fmaxf(cnt, 1.f)) : 0.f;
}

extern "C" void kernel_launch(void* const* d_in, const int* in_sizes, int n_in,
                              void* d_out, int out_size, void* d_ws, size_t ws_size,
                              hipStream_t stream) {
  const float* pred = (const float*)d_in[0];
  const float* targ = (const float*)d_in[1];
  const float* iou  = (const float*)d_in[2];
  float* out = (float*)d_out;
  float* acc = (float*)d_ws;

  const int N = in_sizes[2];
  const int blocks = (N + TILE - 1) / TILE;

  zero_acc<<<1, 1, 0, stream>>>(acc);
  giou3d_kernel<<<blocks, TILE, 0, stream>>>(pred, targ, iou, acc, N);
  finalize<<<1, 1, 0, stream>>>(acc, out);
}